// Channel_attention_15951508537563
// MI455X (gfx1250) — compile-verified
//
#include <hip/hip_runtime.h>
#include <math.h>

typedef __attribute__((ext_vector_type(2))) float v2f;
typedef __attribute__((ext_vector_type(8))) float v8f;

#define CH    128
#define NTOK  4096
#define TILE  16
#define KSTEP (NTOK / TILE)   // 256 key tiles

// One wave per 16-row query tile; 4 waves per block share a double-buffered
// LDS key tile. grid = 256 blocks (4 batches * 64), block = 128 threads.
__launch_bounds__(128)
__global__ void chan_attn_fa(const float* __restrict__ x,
                             const float* __restrict__ beta,
                             float* __restrict__ out) {
  __shared__ float skt[2][TILE * CH];   // 2 x 8 KB key tiles (row-major 16x128)
  __shared__ float sbeta[CH];

  const int tid  = threadIdx.x;
  const int wave = tid >> 5;
  const int lane = tid & 31;
  const int row  = lane & 15;        // M/N index within a 16-wide tile
  const int hi   = lane >> 4;        // which half of the wave

  const int b  = blockIdx.x >> 6;                    // batch
  const int qt = ((blockIdx.x & 63) << 2) + wave;    // query tile id
  const int n0 = qt * TILE;                          // first query row

  const float* xb = x + (size_t)b * NTOK * CH;

  if (tid < CH) sbeta[tid] = beta[tid];

  // ---- Q tile in B-operand layout for V_WMMA_F32_16X16X4_F32 ----
  // q[t] holds { Q[row, 4t+2*hi], Q[row, 4t+2*hi+1] }  (64 VGPRs total)
  v2f q[32];
  {
    const float* qp = xb + (size_t)(n0 + row) * CH + 2 * hi;
#pragma unroll
    for (int t = 0; t < 32; ++t)
      q[t] = *(const v2f*)(qp + 4 * t);
  }

  v8f zero8 = {};
  v8f acc[8];                         // Out^T accumulators: D[M=c', N=m]
#pragma unroll
  for (int jj = 0; jj < 8; ++jj) acc[jj] = zero8;

  float run_max = -INFINITY;
  float row_sum = 0.0f;

  // ---- software pipeline: r holds tile (kt+1) at top of iteration kt ----
  float4 r[4];
  {
    const float4* s0 = (const float4*)xb;                 // tile 0
#pragma unroll
    for (int i = 0; i < 4; ++i) r[i] = s0[tid + i * 128];
    float4* d0 = (float4*)skt[0];
#pragma unroll
    for (int i = 0; i < 4; ++i) d0[tid + i * 128] = r[i];
    const float4* s1 = (const float4*)(xb + (size_t)TILE * CH);  // tile 1
#pragma unroll
    for (int i = 0; i < 4; ++i) r[i] = s1[tid + i * 128];
  }
  __syncthreads();

  for (int kt = 0; kt < KSTEP; ++kt) {
    // stage tile kt+1 into the idle buffer; start fetching tile kt+2
    if (kt + 1 < KSTEP) {
      float4* dst = (float4*)skt[(kt + 1) & 1];
#pragma unroll
      for (int i = 0; i < 4; ++i) dst[tid + i * 128] = r[i];
    }
    if (kt + 2 < KSTEP) {
      const float4* src = (const float4*)(xb + (size_t)(kt + 2) * TILE * CH);
#pragma unroll
      for (int i = 0; i < 4; ++i) r[i] = src[tid + i * 128];
    }
    if (kt + 6 < KSTEP)   // warm WGP$/L2 ahead (global_prefetch_b8)
      __builtin_prefetch(xb + (size_t)(kt + 6) * TILE * CH + tid * 16, 0, 1);

    const float* sk = skt[kt & 1];

    // -------- matmul 1: E'[n,m] = sum_c Kt[n,c] * Q[m,c] --------
    // A chunk t: lane holds Kt[row, 4t+2hi .. +1] (ds_load_b64), B = q[t].
    v8f e = zero8;
    {
      const float* ap = sk + row * CH + 2 * hi;
#pragma unroll
      for (int t = 0; t < 32; ++t) {
        v2f a = *(const v2f*)(ap + 4 * t);
        e = __builtin_amdgcn_wmma_f32_16x16x4_f32(
                false, a, false, q[t], (short)0, e, false, false);
      }
    }
    // e layout: e[v] = E'[n = v + 8*hi, m = row]

    // -------- online softmax over keys n --------
    float tmax = e[0];
#pragma unroll
    for (int v = 1; v < 8; ++v) tmax = fmaxf(tmax, e[v]);
    tmax = fmaxf(tmax, __shfl_xor(tmax, 16, 32));
    float new_max = fmaxf(run_max, tmax);
    float scale   = __expf(run_max - new_max);   // first iter: exp(-inf)=0

    float p[8];
    float tsum = 0.0f;
#pragma unroll
    for (int v = 0; v < 8; ++v) { p[v] = __expf(e[v] - new_max); tsum += p[v]; }
    tsum += __shfl_xor(tsum, 16, 32);
    row_sum = row_sum * scale + tsum;
    run_max = new_max;

    // rescale only when some row max actually moved (wave-uniform branch)
    if (__any(scale < 1.0f)) {
#pragma unroll
      for (int jj = 0; jj < 8; ++jj)
#pragma unroll
        for (int v = 0; v < 8; ++v) acc[jj][v] *= scale;
    }

    // -------- reshape P' (C-layout) into B-operand chunks --------
    // bm[k] vgpr j must hold P'[4k + 2hi + j, m=row]
    float xp[8];
#pragma unroll
    for (int v = 0; v < 8; ++v) xp[v] = __shfl_xor(p[v], 16, 32);
    v2f bm[4];
    bm[0].x = hi ? xp[2] : p[0];  bm[0].y = hi ? xp[3] : p[1];
    bm[1].x = hi ? xp[6] : p[4];  bm[1].y = hi ? xp[7] : p[5];
    bm[2].x = hi ? p[2]  : xp[0]; bm[2].y = hi ? p[3]  : xp[1];
    bm[3].x = hi ? p[6]  : xp[4]; bm[3].y = hi ? p[7]  : xp[5];

    // -------- matmul 2: Out^T[c,m] += sum_n Kt[n,c] * P'[n,m] --------
    // A chunk (k,jj): vgpr0 = Kt[4k+2hi,   16jj+row]
    //                 vgpr1 = Kt[4k+2hi+1, 16jj+row]
#pragma unroll
    for (int k = 0; k < 4; ++k) {
      const float* ab = sk + (4 * k + 2 * hi) * CH + row;
#pragma unroll
      for (int jj = 0; jj < 8; ++jj) {
        v2f a;
        a.x = ab[16 * jj];
        a.y = ab[16 * jj + CH];
        acc[jj] = __builtin_amdgcn_wmma_f32_16x16x4_f32(
                      false, a, false, bm[k], (short)0, acc[jj], false, false);
      }
    }

    if (kt + 1 < KSTEP)
      __syncthreads();   // buf[kt&1] fully consumed; next iter may overwrite
  }

  // -------- epilogue: out = beta[c] * (Out^T / row_sum) + x --------
  const float inv = 1.0f / row_sum;
  float* ob = out + (size_t)b * NTOK * CH;
#pragma unroll
  for (int jj = 0; jj < 8; ++jj) {
#pragma unroll
    for (int v = 0; v < 8; ++v) {
      const int c = 16 * jj + v + 8 * hi;      // Out^T M-index -> channel
      const size_t idx = (size_t)(n0 + row) * CH + c;
      ob[idx] = sbeta[c] * (acc[jj][v] * inv) + xb[idx];
    }
  }
}

extern "C" void kernel_launch(void* const* d_in, const int* in_sizes, int n_in,
                              void* d_out, int out_size, void* d_ws, size_t ws_size,
                              hipStream_t stream) {
  const float* x    = (const float*)d_in[0];   // (4,16,16,16,128) f32
  const float* beta = (const float*)d_in[1];   // (128,) f32
  float* out        = (float*)d_out;           // (4,4096,128) f32
  (void)in_sizes; (void)n_in; (void)out_size; (void)d_ws; (void)ws_size;
  chan_attn_fa<<<dim3(256), dim3(128), 0, stream>>>(x, beta, out);
}